// TemporalCrossCovariance_42090679500988
// MI455X (gfx1250) — compile-verified
//
#include <hip/hip_runtime.h>
#include <stdint.h>

// Temporal cross-covariance diagonal: out[r] = relu((sum(x1*x2) - sum(x1)*sum(x2)/T)/7)
// Pure HBM-bandwidth-bound (256 MiB read -> ~11.5us floor at 23.3 TB/s).
// Uses CDNA5 async global->LDS copies (ASYNCcnt) with double buffering.

#define T_LEN  2048
#define BLOCK  256
#define NWAVES (BLOCK / 32)

// ---- feature probes (device pass only; host pass must not see target builtins)
#if defined(__HIP_DEVICE_COMPILE__) && defined(__has_builtin)
#  if __has_builtin(__builtin_amdgcn_global_load_async_to_lds_b128)
#    define HAVE_ASYNC_BUILTIN 1
#  else
#    define HAVE_ASYNC_BUILTIN 0
#  endif
#  if __has_builtin(__builtin_amdgcn_s_wait_asynccnt)
#    define HAVE_WAIT_BUILTIN 1
#  else
#    define HAVE_WAIT_BUILTIN 0
#  endif
#else
#  define HAVE_ASYNC_BUILTIN 0
#  define HAVE_WAIT_BUILTIN 0
#endif

#if HAVE_WAIT_BUILTIN
#  define WAIT_ASYNC(n) __builtin_amdgcn_s_wait_asynccnt(n)
#else
#  define WAIT_ASYNC(n) asm volatile("s_wait_asynccnt " #n ::: "memory")
#endif

// Builtin signature (from round-1 diagnostic): param0 is v4i in addrspace(1),
// param1 is v4i in addrspace(3), then two immediates (offset, cpol).
typedef __attribute__((ext_vector_type(4))) int v4i;
typedef __attribute__((address_space(1))) v4i GV4;   // global v4i
typedef __attribute__((address_space(3))) v4i LV4;   // LDS v4i

// One wave-wide async B128 copy: 16 bytes/lane, global -> LDS. ASYNCcnt += 1.
__device__ __forceinline__ void async_copy_b128(const float* gsrc, float* ldst) {
  // Low 32 bits of a generic LDS address are the LDS byte offset (ISA 10.2).
  uint32_t lds_off = (uint32_t)(uintptr_t)ldst;
#if HAVE_ASYNC_BUILTIN
  __builtin_amdgcn_global_load_async_to_lds_b128(
      (GV4*)(uintptr_t)gsrc,
      (LV4*)lds_off,
      /*imm offset=*/0, /*cpol=*/0);
#else
  asm volatile("global_load_async_to_lds_b128 %0, %1, off"
               :
               : "v"(lds_off), "v"(gsrc)
               : "memory");
#endif
}

struct __align__(16) Smem {
  float b[2][2][T_LEN];     // [parity][input][t]  : 32 KB double buffer
  float red[NWAVES][4];     // per-wave partials
};

__global__ void __launch_bounds__(BLOCK)
tcc_diag_kernel(const float* __restrict__ x1, const float* __restrict__ x2,
                float* __restrict__ out, int nrows) {
  __shared__ Smem sm;
  const int tid = threadIdx.x;
  const int stride = gridDim.x;
  const int row0 = blockIdx.x;
  if (row0 >= nrows) return;  // uniform per block

  // Issue a full row-pair copy into buffer parity p: 4 async instrs per wave.
  auto issue = [&](int row, int p) {
    const float* g1 = x1 + (size_t)row * T_LEN;
    const float* g2 = x2 + (size_t)row * T_LEN;
    async_copy_b128(g1 + tid * 4,        &sm.b[p][0][tid * 4]);
    async_copy_b128(g1 + 1024 + tid * 4, &sm.b[p][0][1024 + tid * 4]);
    async_copy_b128(g2 + tid * 4,        &sm.b[p][1][tid * 4]);
    async_copy_b128(g2 + 1024 + tid * 4, &sm.b[p][1][1024 + tid * 4]);
  };

  int p = 0;
  issue(row0, 0);
  for (int row = row0; row < nrows; row += stride, p ^= 1) {
    const int next = row + stride;
    if (next < nrows) {
      issue(next, p ^ 1);     // overlap next row's copy with this row's compute
      WAIT_ASYNC(4);          // in-order: current row's 4 copies are done
    } else {
      WAIT_ASYNC(0);
    }
    __syncthreads();          // all waves' copy portions visible

    // 8 floats per thread per input, vectorized LDS reads (ds_load_b128).
    const float4* v1 = reinterpret_cast<const float4*>(sm.b[p][0]);
    const float4* v2 = reinterpret_cast<const float4*>(sm.b[p][1]);
    float4 a0 = v1[tid], a1 = v1[tid + BLOCK];
    float4 b0 = v2[tid], b1 = v2[tid + BLOCK];

    float s1  = ((a0.x + a0.y) + (a0.z + a0.w)) + ((a1.x + a1.y) + (a1.z + a1.w));
    float s2  = ((b0.x + b0.y) + (b0.z + b0.w)) + ((b1.x + b1.y) + (b1.z + b1.w));
    float s12 = a0.x * b0.x;
    s12 = fmaf(a0.y, b0.y, s12);
    s12 = fmaf(a0.z, b0.z, s12);
    s12 = fmaf(a0.w, b0.w, s12);
    s12 = fmaf(a1.x, b1.x, s12);
    s12 = fmaf(a1.y, b1.y, s12);
    s12 = fmaf(a1.z, b1.z, s12);
    s12 = fmaf(a1.w, b1.w, s12);

    // wave32 butterfly reduction
    #pragma unroll
    for (int m = 16; m >= 1; m >>= 1) {
      s1  += __shfl_xor(s1,  m, 32);
      s2  += __shfl_xor(s2,  m, 32);
      s12 += __shfl_xor(s12, m, 32);
    }
    const int wave = tid >> 5;
    if ((tid & 31) == 0) {
      sm.red[wave][0] = s1;
      sm.red[wave][1] = s2;
      sm.red[wave][2] = s12;
    }
    __syncthreads();
    if (tid == 0) {
      float t1 = 0.f, t2 = 0.f, t12 = 0.f;
      #pragma unroll
      for (int w = 0; w < NWAVES; ++w) {
        t1 += sm.red[w][0]; t2 += sm.red[w][1]; t12 += sm.red[w][2];
      }
      float diag = (t12 - t1 * t2 * (1.0f / (float)T_LEN)) * (1.0f / 7.0f);
      out[row] = diag > 0.0f ? diag : 0.0f;
    }
    __syncthreads();          // WAR: buffer p is overwritten next iteration
  }
}

extern "C" void kernel_launch(void* const* d_in, const int* in_sizes, int n_in,
                              void* d_out, int out_size, void* d_ws, size_t ws_size,
                              hipStream_t stream) {
  (void)n_in; (void)d_ws; (void)ws_size; (void)in_sizes;
  const float* x1 = (const float*)d_in[0];
  const float* x2 = (const float*)d_in[1];
  float* out = (float*)d_out;

  const int nrows = out_size;                    // B*C = 16384
  int grid = nrows < 2048 ? nrows : 2048;        // ~8 rows/block for pipelining
  tcc_diag_kernel<<<dim3(grid), dim3(BLOCK), 0, stream>>>(x1, x2, out, nrows);
}